// FusedSOSCascade_18451179504366
// MI455X (gfx1250) — compile-verified
//
#include <hip/hip_runtime.h>

// CDNA5 (gfx1250) wave32 block-IIR SOS cascade.
//  - chunk-parallel with warm-up overlap (stable biquads => geometric state decay)
//  - per 16-sample tile: y = H * x_ext(20) + [p q] * [y1;y2]  via 6x V_WMMA_F32_16X16X4_F32
//  - async global<->LDS staging (GLOBAL_LOAD_ASYNC_TO_LDS_B128 / STORE_ASYNC_FROM_LDS_B128)
//  - state WMMA placed last so ds_bpermute latency overlaps the 5 H-WMMAs

typedef __attribute__((ext_vector_type(2))) float v2f;
typedef __attribute__((ext_vector_type(8))) float v8f;

#define BODY    512                   // samples produced per chunk (multiple of 128)
#define HWARM   256                   // warm-up samples re-run from zero state
#define MAXLEN  (BODY + HWARM)        // 768
#define FRONT   4                     // front pad: idx2 = x[-2], idx3 = x[-1]
#define RS      (FRONT + MAXLEN + 8)  // 780; 780 % 64 == 12 -> 16 rows hit distinct LDS banks
#define GMAX    18                    // impulse response length kept (16 + 2)

static __device__ __forceinline__ v8f wmma_16x16x4_f32(v2f a, v2f b, v8f c) {
  // 8-arg pattern: (neg_a, A, neg_b, B, c_mod, C, reuse_a, reuse_b)
  return __builtin_amdgcn_wmma_f32_16x16x4_f32(false, a, false, b, (short)0, c,
                                               false, false);
}

// Async DMA: global -> LDS, 16 bytes per lane, tracked by ASYNCcnt (no VGPR data path).
static __device__ __forceinline__ void async_load_b128(unsigned lds_addr, const void* gaddr) {
  asm volatile("global_load_async_to_lds_b128 %0, %1, off"
               :: "v"(lds_addr), "v"(gaddr) : "memory");
}
// Async DMA: LDS -> global, 16 bytes per lane.
static __device__ __forceinline__ void async_store_b128(void* gaddr, unsigned lds_addr) {
  asm volatile("global_store_async_from_lds_b128 %0, %1, off"
               :: "v"(gaddr), "v"(lds_addr) : "memory");
}
static __device__ __forceinline__ void wait_async0() {
  asm volatile("s_wait_asynccnt 0x0" ::: "memory");
}
static __device__ __forceinline__ void wait_ds0() {
  asm volatile("s_wait_dscnt 0x0" ::: "memory");
}

__global__ __launch_bounds__(32)
void sos_wmma_kernel(const float* __restrict__ x,
                     const float* __restrict__ sos,
                     const float* __restrict__ stx,
                     const float* __restrict__ sty,
                     float* __restrict__ out,
                     int T, int C, int K)
{
  __shared__ float buf[16 * RS];   // 16 channels x padded chunk, ~50 KB
  __shared__ float gsh[GMAX];      // full-biquad impulse response g[n]
  __shared__ float hsh[GMAX];      // AR-part impulse response h[n]

  const int lane = threadIdx.x;    // 0..31 (wave32)
  const int m    = lane & 15;      // row (time) for A, channel (column) for B/D
  const int half = lane >> 4;      // lane half selects K/M sub-block
  const int c0   = blockIdx.y * 16;
  const int bodyStart = blockIdx.x * BODY;
  const int loadStart = (bodyStart >= HWARM) ? (bodyStart - HWARM) : 0;
  const int warm = bodyStart - loadStart;
  const int len  = warm + BODY;            // <= MAXLEN, multiple of 128
  const bool first = (bodyStart == 0);

  // LDS byte offset of buf (low 32 bits of the shared-aperture address = LDS offset)
  const unsigned ldsBuf = (unsigned)(uintptr_t)(void*)buf;

  // ---- stage section-0 input chunk into LDS with async DMA ----
  for (int c = 0; c < 16; ++c) {
    const float* src = x + (size_t)(c0 + c) * T + loadStart;
    const unsigned dstRow = ldsBuf + (unsigned)((c * RS + FRONT) * 4);
    for (int t = lane * 4; t < len; t += 128)
      async_load_b128(dstRow + (unsigned)(t * 4), src + t);
  }
  wait_async0();
  if (lane < 16) {                          // back pad (read under zero H-columns)
    buf[lane * RS + FRONT + len]     = 0.f;
    buf[lane * RS + FRONT + len + 1] = 0.f;
  }
  __syncthreads();

  for (int s = 0; s < K; ++s) {
    const float b0 = sos[s*6+0], b1 = sos[s*6+1], b2 = sos[s*6+2];
    const float a1 = sos[s*6+4], a2 = sos[s*6+5];

    // input history of this section at chunk start (exact for chunk 0, zero otherwise)
    if (lane < 16) {
      float xm1 = 0.f, xm2 = 0.f;
      if (first) {
        xm1 = stx[((size_t)s * C + c0 + lane) * 2 + 0];
        xm2 = stx[((size_t)s * C + c0 + lane) * 2 + 1];
      }
      buf[lane * RS + FRONT - 1] = xm1;   // t = -1
      buf[lane * RS + FRONT - 2] = xm2;   // t = -2
    }
    float y1 = 0.f, y2 = 0.f;             // output history (per channel = lane&15)
    if (first) {
      y1 = sty[((size_t)s * C + c0 + m) * 2 + 0];
      y2 = sty[((size_t)s * C + c0 + m) * 2 + 1];
    }

    // impulse responses: h (AR part), g = conv(b, h) (full biquad)
    if (lane == 0) {
      float hm1 = 0.f, hm2 = 0.f;
      for (int n = 0; n < GMAX; ++n) {
        float hn = ((n == 0) ? 1.f : 0.f) - a1 * hm1 - a2 * hm2;
        gsh[n] = b0 * hn + b1 * hm1 + b2 * hm2;
        hsh[n] = hn;
        hm2 = hm1; hm1 = hn;
      }
    }
    __syncthreads();

    // A-operands (constant per section): 5 slices of H(16x20) + state slice [p q 0 0]
    // A 16x4 layout: lanes0-15 hold K = 2*half+{0,1} in vgpr {x,y}; M = lane&15.
    v2f aH[5], apq;
#pragma unroll
    for (int j = 0; j < 5; ++j) {
      const int col = 4 * j + 2 * half;
      const int i0 = m + 2 - col;        // H[m][col]   = g[m+2-col]
      const int i1 = m + 1 - col;        // H[m][col+1] = g[m+1-col]
      float g0 = gsh[max(0, min(i0, GMAX - 1))];
      float g1 = gsh[max(0, min(i1, GMAX - 1))];
      if (i0 < 0 || i0 >= GMAX) g0 = 0.f;
      if (i1 < 0 || i1 >= GMAX) g1 = 0.f;
      aH[j].x = g0; aH[j].y = g1;
    }
    apq.x = half ? 0.f : hsh[m + 1];       // p[m] = h[m+1]
    apq.y = half ? 0.f : (-a2 * hsh[m]);   // q[m] = -a2*h[m]

    // ---- serial tile loop: 16 time-samples per step on the matrix pipe ----
    for (int t0 = 0; t0 < len; t0 += 16) {
      v8f acc = {};
      // B slices: channel = lane&15, times = t0-2 + 4j + 2*half + {0,1}
      const float* brow = buf + m * RS + (FRONT - 2) + t0 + 2 * half;
#pragma unroll
      for (int j = 0; j < 5; ++j) {
        v2f b;
        b.x = brow[4 * j];
        b.y = brow[4 * j + 1];
        acc = wmma_16x16x4_f32(aH[j], b, acc);
      }
      // state correction last: ds_bpermute results had 5 WMMAs to land
      v2f bpq; bpq.x = y1; bpq.y = y2;
      acc = wmma_16x16x4_f32(apq, bpq, acc);

      // write tile back (input of next section / final output)
      float* drow = buf + m * RS + FRONT + t0 + 8 * half;
#pragma unroll
      for (int r = 0; r < 8; ++r) drow[r] = acc[r];

      // next-tile output history straight from the D accumulator:
      // y[t0+15] = vgpr7 upper half, y[t0+14] = vgpr6 upper half
      y1 = __shfl(acc[7], m + 16, 32);
      y2 = __shfl(acc[6], m + 16, 32);
    }
    __syncthreads();
  }

  // ---- emit body region with async LDS -> global DMA ----
  wait_ds0();                       // tile ds_stores complete before DMA engine reads LDS
  for (int c = 0; c < 16; ++c) {
    float* dst = out + (size_t)(c0 + c) * T + bodyStart;
    const unsigned srcRow = ldsBuf + (unsigned)((c * RS + FRONT + warm) * 4);
    for (int t = lane * 4; t < BODY; t += 128)
      async_store_b128(dst + t, srcRow + (unsigned)(t * 4));
  }
  wait_async0();                    // drain before wave exit (S_ENDPGM would also wait-idle)
}

extern "C" void kernel_launch(void* const* d_in, const int* in_sizes, int n_in,
                              void* d_out, int out_size, void* d_ws, size_t ws_size,
                              hipStream_t stream) {
  const float* x   = (const float*)d_in[0];
  const float* sos = (const float*)d_in[1];
  const float* stx = (const float*)d_in[2];
  const float* sty = (const float*)d_in[3];
  float* out = (float*)d_out;

  const int K = in_sizes[1] / 6;           // 8 sections
  const int C = in_sizes[2] / (K * 2);     // 64 channels
  const int T = in_sizes[0] / C;           // 262144 (multiple of BODY)

  dim3 grid(T / BODY, C / 16);             // 512 chunks x 4 channel groups
  sos_wmma_kernel<<<grid, 32, 0, stream>>>(x, sos, stx, sty, out, T, C, K);
}